// RoIPool3d_46084999086895
// MI455X (gfx1250) — compile-verified
//
#include <hip/hip_runtime.h>
#include <cfloat>
#include <cmath>

#define OUTV 5
#define NVOX 125          // 5^3
#define NB   2
#define NP   8192
#define NR   64
#define NC   64
#define TP   1024         // points per LDS tile
#define NT   (NP / TP)    // 8 tiles
#define BLK  256
#define PSTR 65           // padded pool row stride (conflict-free on 64 banks)

// Low 32 bits of a flat LDS pointer == LDS byte offset (ISA aperture rule).
__device__ __forceinline__ unsigned lds_off(const void* p) {
  return (unsigned)(unsigned long long)p;
}

// CDNA5 LDS float atomic max (IEEE maximumNumber) — invisible to compiler's
// DScnt tracking, so we issue explicit s_wait_dscnt before the final barrier.
__device__ __forceinline__ void lds_max_f32(float* p, float v) {
  asm volatile("ds_max_num_f32 %0, %1" :: "v"(lds_off(p)), "v"(v) : "memory");
}

// CDNA5 async VMEM->LDS copy (16B per lane), tracked by ASYNCcnt.
__device__ __forceinline__ void async_ld_b128(unsigned ldsoff, unsigned long long gaddr) {
  asm volatile("global_load_async_to_lds_b128 %0, %1, off"
               :: "v"(ldsoff), "v"(gaddr) : "memory");
}

__global__ __launch_bounds__(BLK) void roipool3d_kernel(
    const float* __restrict__ points,   // [B, P, 3]
    const float* __restrict__ feats,    // [B, C, P]
    const float* __restrict__ rois,     // [B, R, 7]
    float* __restrict__ out)            // [B*R, C, 125]
{
  __shared__ __align__(16) float s_pts[2][TP * 3];   // 24 KB double buffer
  __shared__ float s_pool[NVOX * PSTR];              // 32.5 KB pooled max

  const int bid  = blockIdx.x;          // bid = b*NR + r
  const int b    = bid / NR;
  const int tid  = threadIdx.x;
  const int lane = tid & 31;
  const int wv   = tid >> 5;

  // ROI parameters (uniform per block)
  const float* rp  = rois + (size_t)bid * 7;
  const float  cx  = rp[0], cy = rp[1];
  const float  hx  = 0.5f * rp[3], hy = 0.5f * rp[4], hz = 0.5f * rp[5];
  const float  cz  = rp[2] + hz;                       // geometric z center
  const float  vsx = rp[3] / (float)OUTV;
  const float  vsy = rp[4] / (float)OUTV;
  const float  vsz = rp[5] / (float)OUTV;
  const float  cth = cosf(rp[6]), sth = sinf(rp[6]);

  const float* pbase = points + (size_t)b * NP * 3;
  const float* fbase = feats  + (size_t)b * NC * NP;

  // Init pooled maxima to -FLT_MAX (empty sentinel -> 0 at write-out).
  for (int i = tid; i < NVOX * PSTR; i += BLK) s_pool[i] = -FLT_MAX;

  // Prefetch tile 0 (12 KB = 3 x 16B chunks per thread), async into LDS.
  {
    const unsigned l0 = lds_off(&s_pts[0][0]);
    const unsigned long long g0 = (unsigned long long)(const void*)pbase;
#pragma unroll
    for (int k = 0; k < 3; ++k) {
      const unsigned off = (unsigned)tid * 16u + (unsigned)k * (BLK * 16u);
      async_ld_b128(l0 + off, g0 + off);
    }
  }
  __syncthreads();  // s_pool init visible before any ds_max

  for (int t = 0; t < NT; ++t) {
    const int cur = t & 1;
    // Kick next tile into the other buffer, then wait for current tile's
    // 3 in-flight async ops (in-order completion => asynccnt <= 3 suffices).
    if (t + 1 < NT) {
      const unsigned l0 = lds_off(&s_pts[cur ^ 1][0]);
      const unsigned long long g0 =
          (unsigned long long)(const void*)(pbase + (size_t)(t + 1) * TP * 3);
#pragma unroll
      for (int k = 0; k < 3; ++k) {
        const unsigned off = (unsigned)tid * 16u + (unsigned)k * (BLK * 16u);
        async_ld_b128(l0 + off, g0 + off);
      }
      asm volatile("s_wait_asynccnt 0x3" ::: "memory");
    } else {
      asm volatile("s_wait_asynccnt 0x0" ::: "memory");
    }
    __syncthreads();  // all waves' current tile landed in LDS

    const float* sp = s_pts[cur];
#pragma unroll
    for (int k = 0; k < TP / BLK; ++k) {
      const int   j  = k * BLK + tid;
      const float x  = sp[j * 3 + 0];
      const float y  = sp[j * 3 + 1];
      const float z  = sp[j * 3 + 2];
      const float sx = x - cx, sy = y - cy, sz = z - cz;
      const float lx =  sx * cth + sy * sth;   // rotate world -> box (-ry)
      const float ly =  sy * cth - sx * sth;
      const float lz =  sz;
      const bool inside = (fabsf(lx) < hx) & (fabsf(ly) < hy) & (fabsf(lz) < hz);

      int vx = (int)floorf((lx + hx) / vsx);
      int vy = (int)floorf((ly + hy) / vsy);
      int vz = (int)floorf((lz + hz) / vsz);
      vx = min(max(vx, 0), OUTV - 1);
      vy = min(max(vy, 0), OUTV - 1);
      vz = min(max(vz, 0), OUTV - 1);
      const int vid = (vx * OUTV + vy) * OUTV + vz;

      // Inside hits are rare (~10/ROI): whole wave services each hit,
      // 2 channels per lane, conflict-free ds_max_num_f32 scatter.
      unsigned  mm    = __builtin_amdgcn_ballot_w32(inside);
      const int pidx0 = t * TP + k * BLK + wv * 32;
      while (mm) {
        const int L = __builtin_ctz(mm);
        mm &= mm - 1u;
        const int   vL   = __builtin_amdgcn_readlane(vid, L);  // L uniform
        const int   pidx = pidx0 + L;
        const float v0   = fbase[(size_t)lane * NP + pidx];
        const float v1   = fbase[(size_t)(lane + 32) * NP + pidx];
        lds_max_f32(&s_pool[vL * PSTR + lane],      v0);
        lds_max_f32(&s_pool[vL * PSTR + lane + 32], v1);
      }
    }
    __syncthreads();  // everyone done reading buf[cur] before it is refilled
  }

  // Our inline-asm DS atomics are untracked by the compiler: drain DScnt
  // explicitly, then barrier, then read the pooled results.
  asm volatile("s_wait_dscnt 0x0" ::: "memory");
  __syncthreads();

  float* ob = out + (size_t)bid * (NC * NVOX);
  for (int i = tid; i < NC * NVOX; i += BLK) {   // coalesced [C][125] write
    const int   c   = i / NVOX;
    const int   v   = i - c * NVOX;
    const float val = s_pool[v * PSTR + c];      // stride-65: conflict-free
    ob[i] = (val == -FLT_MAX) ? 0.0f : val;
  }
}

extern "C" void kernel_launch(void* const* d_in, const int* in_sizes, int n_in,
                              void* d_out, int out_size, void* d_ws, size_t ws_size,
                              hipStream_t stream) {
  (void)in_sizes; (void)n_in; (void)out_size; (void)d_ws; (void)ws_size;
  const float* points = (const float*)d_in[0];  // [B,P,3]
  const float* feats  = (const float*)d_in[1];  // [B,C,P]
  const float* rois   = (const float*)d_in[2];  // [B,R,7]
  float*       out    = (float*)d_out;          // [B*R, C, 125]
  roipool3d_kernel<<<dim3(NB * NR), dim3(BLK), 0, stream>>>(points, feats, rois, out);
}